// JumpSelfAttention_43903155699724
// MI455X (gfx1250) — compile-verified
//
#include <hip/hip_runtime.h>
#include <hip/hip_bf16.h>

typedef float v2f __attribute__((ext_vector_type(2)));
typedef float v8f __attribute__((ext_vector_type(8)));

#define L_SEQ   512
#define D_MODEL 512
#define H_HEADS 8
#define HD_DIM  64
#define B_BATCH 4
#define BH      (B_BATCH * H_HEADS)
#define RHO_TH  0.1f

// ---------------------------------------------------------------------------
// Batched GEMM via V_WMMA_F32_16X16X4_F32.
// C[b,h] = alpha * A[b,h] (MxK, row-major lda) * op(B[b,h]) (KxN; row-major ldb,
//          or NxK row-major if TRANSB).
// Block = 128 threads (4 waves). Each wave computes a 16x64 strip of C
// (4 accumulators, A fragment reused 4x per K-step). A fragments (and TRANSB
// B fragments) are 8B-aligned contiguous pairs -> vector v2f loads (b64).
// Fragment layouts per CDNA5 ISA 7.12.2:
//   A 16x4 : lanes 0-15 hold (M=lane, K=k,k+1); lanes 16-31 hold (M, K=k+2,k+3)
//   B 4x16 : lanes 0-15 hold (K=k,k+1, N=lane); lanes 16-31 hold (K=k+2,k+3, N)
//   C 16x16: VGPR r -> lanes 0-15: (M=r, N=lane); lanes 16-31: (M=r+8, N=lane-16)
// ---------------------------------------------------------------------------
template<bool TRANSB>
__global__ __launch_bounds__(128)
void wmma_gemm(const float* __restrict__ Abase, const float* __restrict__ Bbase,
               float* __restrict__ Cbase,
               int M, int N, int K, int lda, int ldb, int ldc,
               long sAb, long sAh, long sBb, long sBh, long sCb, long sCh,
               int Hn, float alpha)
{
    const int z = blockIdx.z;
    const int b = z / Hn;
    const int h = z % Hn;
    const float* __restrict__ A = Abase + (long)b * sAb + (long)h * sAh;
    const float* __restrict__ B = Bbase + (long)b * sBb + (long)h * sBh;
    float* __restrict__ C       = Cbase + (long)b * sCb + (long)h * sCh;

    const int wave  = threadIdx.x >> 5;
    const int lane  = threadIdx.x & 31;
    const int strip = blockIdx.x * 4 + wave;
    const int j0    = strip * 64;
    if (j0 >= N) return;                 // whole-wave uniform: EXEC stays all-1s
    const int i0 = blockIdx.y * 16;

    const int lm = lane & 15;            // M / N index within fragment
    const int lk = (lane >> 4) << 1;     // K sub-offset: 0 or 2

    // Warm L2/WGP$ for the panels this wave will stream (global_prefetch_b8).
    __builtin_prefetch(A + (long)(i0 + lm) * lda + lk, 0, 3);
    if (TRANSB) __builtin_prefetch(B + (long)(j0 + lm) * ldb + lk, 0, 3);
    else        __builtin_prefetch(B + (long)lk * ldb + j0 + lm, 0, 3);

    v8f acc[4] = {v8f{}, v8f{}, v8f{}, v8f{}};

    for (int k = 0; k < K; k += 4) {
        const int kk = k + lk;
        const v2f a = *(const v2f*)(A + (long)(i0 + lm) * lda + kk);

        #pragma unroll
        for (int t = 0; t < 4; ++t) {
            const int n = j0 + t * 16 + lm;
            v2f bf;
            if (TRANSB) {
                bf = *(const v2f*)(B + (long)n * ldb + kk);   // contiguous pair
            } else {
                bf.x = B[(long)kk * ldb + n];
                bf.y = B[(long)(kk + 1) * ldb + n];
            }
            acc[t] = __builtin_amdgcn_wmma_f32_16x16x4_f32(
                false, a, false, bf, (short)0, acc[t], false, false);
        }
    }

    const int mrow = i0 + ((lane >> 4) << 3);
    #pragma unroll
    for (int t = 0; t < 4; ++t) {
        const int n = j0 + t * 16 + lm;
        #pragma unroll
        for (int r = 0; r < 8; ++r) {
            C[(long)(mrow + r) * ldc + n] = alpha * acc[t][r];
        }
    }
}

// ---------------------------------------------------------------------------
// Adjacency build (thresholded — not expressible as matmul, VALU path):
// A[i,k] = (1/L) * sum_j  p*(p>RHO),  p = S[i,j]*S[k,j]/HD,  diag zeroed.
// 32x32 output tile per block; S panels staged into LDS via the CDNA5 async
// global->LDS path (global_load_async_to_lds_b32, ASYNCcnt), fenced with
// s_wait_asynccnt before the workgroup barrier. Row pitch 65 floats ->
// Sk[lane][j] is bank-conflict-free, Si[row][j] is a broadcast.
// ---------------------------------------------------------------------------
__global__ __launch_bounds__(256)
void build_adjacency(const float* __restrict__ S, float* __restrict__ A)
{
    const int z = blockIdx.z;
    const float* __restrict__ Sz = S + (long)z * L_SEQ * L_SEQ;
    float* __restrict__ Az       = A + (long)z * L_SEQ * L_SEQ;

    __shared__ float Si[32][65];
    __shared__ float Sk[32][65];

    const int i0 = blockIdx.y * 32;
    const int k0 = blockIdx.x * 32;
    const int t  = threadIdx.x;
    const int kc = t & 31;       // output column within tile (== lane)
    const int ir = t >> 5;       // output row group 0..7

    float acc[4] = {0.f, 0.f, 0.f, 0.f};
    const float invHD = 1.0f / (float)HD_DIM;

    for (int j0 = 0; j0 < L_SEQ; j0 += 64) {
        // Async global->LDS copy of two 32x64 panels (8KB each).
        #pragma unroll
        for (int u = 0; u < 8; ++u) {
            const int idx = t + u * 256;       // 0..2047
            const int row = idx >> 6;
            const int col = idx & 63;
            const float* gsi = Sz + (long)(i0 + row) * L_SEQ + j0 + col;
            const float* gsk = Sz + (long)(k0 + row) * L_SEQ + j0 + col;
            // Flat->LDS aperture mapping: LDS offset is the pointer's low 32 bits
            unsigned si_off = (unsigned)(uintptr_t)&Si[row][col];
            unsigned sk_off = (unsigned)(uintptr_t)&Sk[row][col];
            asm volatile("global_load_async_to_lds_b32 %0, %1, off"
                         :: "v"(si_off), "v"(gsi) : "memory");
            asm volatile("global_load_async_to_lds_b32 %0, %1, off"
                         :: "v"(sk_off), "v"(gsk) : "memory");
        }
        asm volatile("s_wait_asynccnt 0x0" ::: "memory");
        __syncthreads();

        #pragma unroll 4
        for (int j = 0; j < 64; ++j) {
            const float skv = Sk[kc][j] * invHD;
            #pragma unroll
            for (int r = 0; r < 4; ++r) {
                const float p = Si[ir + r * 8][j] * skv;
                acc[r] += (p > RHO_TH) ? p : 0.0f;
            }
        }
        __syncthreads();   // protect LDS from next iteration's async writes
    }

    const float invL = 1.0f / (float)L_SEQ;
    #pragma unroll
    for (int r = 0; r < 4; ++r) {
        const int ig = i0 + ir + r * 8;
        const int kg = k0 + kc;
        Az[(long)ig * L_SEQ + kg] = (ig == kg) ? 0.0f : acc[r] * invL;
    }
}

// D^{-1/2} per row of A_tilde = A + I.  One wave (32 lanes) per row.
__global__ __launch_bounds__(256)
void compute_dinv(const float* __restrict__ A, float* __restrict__ dinv)
{
    const int z    = blockIdx.y;
    const int row  = blockIdx.x * 8 + (threadIdx.x >> 5);
    const int lane = threadIdx.x & 31;
    const float* __restrict__ Ar = A + ((long)z * L_SEQ + row) * L_SEQ;

    float s = 0.f;
    #pragma unroll
    for (int j = lane; j < L_SEQ; j += 32) s += Ar[j];
    #pragma unroll
    for (int off = 16; off > 0; off >>= 1) s += __shfl_xor(s, off, 32);
    s += 1.0f;  // diagonal contribution of A_tilde
    if (lane == 0)
        dinv[(long)z * L_SEQ + row] = rsqrtf(fmaxf(s, 1e-6f));
}

// In-place: A_hat[i,k] = dinv[i] * (A[i,k] + (i==k)) * dinv[k]
__global__ __launch_bounds__(256)
void normalize_adjacency(float* __restrict__ A, const float* __restrict__ dinv)
{
    const long idx = (long)blockIdx.x * 256 + threadIdx.x;   // over BH*L*L
    const int z   = (int)(idx / ((long)L_SEQ * L_SEQ));
    const int rem = (int)(idx % ((long)L_SEQ * L_SEQ));
    const int i   = rem / L_SEQ;
    const int k   = rem % L_SEQ;
    const float v = A[idx] + ((i == k) ? 1.0f : 0.0f);
    A[idx] = v * dinv[(long)z * L_SEQ + i] * dinv[(long)z * L_SEQ + k];
}

// Row softmax in place. One wave per row, 16 elements per lane in registers.
__global__ __launch_bounds__(256)
void softmax_rows(float* __restrict__ S)
{
    const int z    = blockIdx.y;
    const int row  = blockIdx.x * 8 + (threadIdx.x >> 5);
    const int lane = threadIdx.x & 31;
    float* __restrict__ r = S + ((long)z * L_SEQ + row) * L_SEQ;

    float v[16];
    float m = -INFINITY;
    #pragma unroll
    for (int u = 0; u < 16; ++u) { v[u] = r[lane + u * 32]; m = fmaxf(m, v[u]); }
    #pragma unroll
    for (int off = 16; off > 0; off >>= 1) m = fmaxf(m, __shfl_xor(m, off, 32));
    float s = 0.f;
    #pragma unroll
    for (int u = 0; u < 16; ++u) { v[u] = __expf(v[u] - m); s += v[u]; }
    #pragma unroll
    for (int off = 16; off > 0; off >>= 1) s += __shfl_xor(s, off, 32);
    const float inv = 1.0f / s;
    #pragma unroll
    for (int u = 0; u < 16; ++u) r[lane + u * 32] = v[u] * inv;
}

// ---------------------------------------------------------------------------
extern "C" void kernel_launch(void* const* d_in, const int* in_sizes, int n_in,
                              void* d_out, int out_size, void* d_ws, size_t ws_size,
                              hipStream_t stream)
{
    const float* x  = (const float*)d_in[0];
    const float* Wq = (const float*)d_in[1];
    const float* Wk = (const float*)d_in[2];
    const float* Wv = (const float*)d_in[3];
    const float* Wo = (const float*)d_in[4];
    float* out = (float*)d_out;

    char* ws = (char*)d_ws;
    float* Q    = (float*)(ws);                    //  4 MB  [B,L,H*HD]
    float* Kb   = (float*)(ws + (4LL  << 20));     //  4 MB
    float* V    = (float*)(ws + (8LL  << 20));     //  4 MB
    float* S    = (float*)(ws + (12LL << 20));     // 32 MB  [B,H,L,L]
    float* Adj  = (float*)(ws + (44LL << 20));     // 32 MB
    float* T1   = (float*)(ws + (76LL << 20));     // 32 MB
    float* dinv = (float*)(ws + (108LL << 20));    // 64 KB  [B,H,L]

    const long sQb = (long)L_SEQ * D_MODEL;        // per-batch stride in [B,L,H*HD]
    const long sQh = HD_DIM;                       // per-head column offset
    const long sSb = (long)H_HEADS * L_SEQ * L_SEQ;
    const long sSh = (long)L_SEQ * L_SEQ;

    const dim3 blk(128);
    const int ML = B_BATCH * L_SEQ;                // 2048

    // 1) Projections: [2048,512] @ [512,512]
    {
        dim3 grid(2, ML / 16, 1);
        wmma_gemm<false><<<grid, blk, 0, stream>>>(x, Wq, Q,  ML, 512, 512, 512, 512, 512,
                                                   0, 0, 0, 0, 0, 0, 1, 1.0f);
        wmma_gemm<false><<<grid, blk, 0, stream>>>(x, Wk, Kb, ML, 512, 512, 512, 512, 512,
                                                   0, 0, 0, 0, 0, 0, 1, 1.0f);
        wmma_gemm<false><<<grid, blk, 0, stream>>>(x, Wv, V,  ML, 512, 512, 512, 512, 512,
                                                   0, 0, 0, 0, 0, 0, 1, 1.0f);
    }
    // 2) S = Q K^T per head: [512,512] = [512,64] @ [512,64]^T
    {
        dim3 grid(2, L_SEQ / 16, BH);
        wmma_gemm<true><<<grid, blk, 0, stream>>>(Q, Kb, S, L_SEQ, L_SEQ, HD_DIM,
                                                  512, 512, 512,
                                                  sQb, sQh, sQb, sQh, sSb, sSh,
                                                  H_HEADS, 1.0f);
    }
    // 3) Adjacency (VALU, async-LDS staged)
    {
        dim3 grid(L_SEQ / 32, L_SEQ / 32, BH);
        build_adjacency<<<grid, 256, 0, stream>>>(S, Adj);
    }
    // 4) Symmetric normalization (in place)
    {
        dim3 grid1(L_SEQ / 8, BH);
        compute_dinv<<<grid1, 256, 0, stream>>>(Adj, dinv);
        long n = (long)BH * L_SEQ * L_SEQ;
        normalize_adjacency<<<(unsigned)(n / 256), 256, 0, stream>>>(Adj, dinv);
    }
    // 5) T1 = A_hat @ S ; S_jump = T1 @ A_hat^T / sqrt(HD)  (into S buffer)
    {
        dim3 grid(2, L_SEQ / 16, BH);
        wmma_gemm<false><<<grid, blk, 0, stream>>>(Adj, S, T1, L_SEQ, L_SEQ, L_SEQ,
                                                   512, 512, 512,
                                                   sSb, sSh, sSb, sSh, sSb, sSh,
                                                   H_HEADS, 1.0f);
        wmma_gemm<true><<<grid, blk, 0, stream>>>(T1, Adj, S, L_SEQ, L_SEQ, L_SEQ,
                                                  512, 512, 512,
                                                  sSb, sSh, sSb, sSh, sSb, sSh,
                                                  H_HEADS, 0.125f);
    }
    // 6) Softmax rows (in place on S)
    {
        dim3 grid(L_SEQ / 8, BH);
        softmax_rows<<<grid, 256, 0, stream>>>(S);
    }
    // 7) attn @ V  -> concat layout [B,L,H*HD] (reuse Q buffer)
    {
        dim3 grid(1, L_SEQ / 16, BH);
        wmma_gemm<false><<<grid, blk, 0, stream>>>(S, V, Q, L_SEQ, HD_DIM, L_SEQ,
                                                   512, 512, 512,
                                                   sSb, sSh, sQb, sQh, sQb, sQh,
                                                   H_HEADS, 1.0f);
    }
    // 8) out = concat @ Wo : [2048,512] @ [512,512]
    {
        dim3 grid(2, ML / 16, 1);
        wmma_gemm<false><<<grid, blk, 0, stream>>>(Q, Wo, out, ML, 512, 512, 512, 512, 512,
                                                   0, 0, 0, 0, 0, 0, 1, 1.0f);
    }
}